// YOLOv5_36077725286807
// MI455X (gfx1250) — compile-verified
//
#include <hip/hip_runtime.h>
#include <hip/hip_bf16.h>
#include <stdint.h>

// ---------------- constants ----------------
#define TOPK_    1000
#define NCAND    3000           // 3 levels * 1000
#define NCLS_    80
#define NBINS    8192
#define BIN_SHIFT 17            // score in [0,1): bits>>17 < 8128 < 8192
#define CAP_     8192           // per-level candidate capacity (pow2 for bitonic)
#define NWORDS   94             // ceil(3000/32)
#define NROWPAD  3008           // 188 * 16
#define CONF_T   0.01f
#define NMS_T    0.5f

typedef __attribute__((ext_vector_type(2))) float v2f;
typedef __attribute__((ext_vector_type(8))) float v8f;
typedef unsigned long long u64;
typedef unsigned int u32;

#if defined(__HIP_DEVICE_COMPILE__) && __has_builtin(__builtin_amdgcn_wmma_f32_16x16x4_f32)
#define HAVE_WMMA_F32X4 1
#endif

#if defined(__HIP_DEVICE_COMPILE__) && \
    __has_builtin(__builtin_amdgcn_global_load_async_to_lds_b64) && \
    __has_builtin(__builtin_amdgcn_global_load_async_to_lds_b128) && \
    __has_builtin(__builtin_amdgcn_s_wait_asynccnt)
#define HAVE_ASYNC_LDS 1
// builtin params are (global int-vector*, lds int-vector*, imm offset, imm cpol)
typedef int vi2 __attribute__((vector_size(8)));
typedef int vi4 __attribute__((vector_size(16)));
#define GV2(p) ((__attribute__((address_space(1))) vi2*)(p))
#define LV2(p) ((__attribute__((address_space(3))) vi2*)(p))
#define GV4(p) ((__attribute__((address_space(1))) vi4*)(p))
#define LV4(p) ((__attribute__((address_space(3))) vi4*)(p))
#endif

__device__ __forceinline__ float fast_sig(float x) { return 1.0f / (1.0f + __expf(-x)); }
__device__ __forceinline__ float ref_sig (float x) { return 1.0f / (1.0f + expf(-x)); }
__device__ __forceinline__ int   imin    (int a, int b) { return a < b ? a : b; }

// ---------------- 0: zero the select/compact scratch ----------------
__global__ void k_zero(u32* __restrict__ p, int n) {
    int stride = gridDim.x * blockDim.x;
    for (int i = blockIdx.x * blockDim.x + threadIdx.x; i < n; i += stride) p[i] = 0u;
}

// ---------------- 1: streaming score histogram (pass 1) ----------------
// n4 = M*80/4 float4 elements of cls. 80 % 4 == 0 so one float4 never crosses a row.
__global__ void __launch_bounds__(256) k_hist(const float* __restrict__ obj,
                                              const float* __restrict__ cls,
                                              int n4, u32* __restrict__ hist) {
    __shared__ u32 lh[NBINS];
    for (int t = threadIdx.x; t < NBINS; t += blockDim.x) lh[t] = 0u;
    __syncthreads();
    int stride = gridDim.x * blockDim.x;
    for (int i = blockIdx.x * blockDim.x + threadIdx.x; i < n4; i += stride) {
        __builtin_prefetch(reinterpret_cast<const char*>(cls) + 16ull * ((unsigned)i + (unsigned)stride), 0, 1);
        int row = i / 20;                                   // 20 float4 per anchor row
        float os = fast_sig(obj[row]);
        float m = (os > CONF_T) ? os : 0.0f;                // mask -> exact 0.0 like reference
        float4 c = reinterpret_cast<const float4*>(cls)[i];
        float s0 = m * fast_sig(c.x);
        float s1 = m * fast_sig(c.y);
        float s2 = m * fast_sig(c.z);
        float s3 = m * fast_sig(c.w);
        atomicAdd(&lh[__float_as_uint(s0) >> BIN_SHIFT], 1u);
        atomicAdd(&lh[__float_as_uint(s1) >> BIN_SHIFT], 1u);
        atomicAdd(&lh[__float_as_uint(s2) >> BIN_SHIFT], 1u);
        atomicAdd(&lh[__float_as_uint(s3) >> BIN_SHIFT], 1u);
    }
    __syncthreads();
    for (int t = threadIdx.x; t < NBINS; t += blockDim.x)
        if (lh[t]) atomicAdd(&hist[t], lh[t]);
}

// ---------------- 2: radix-select cutoff bin per level ----------------
__global__ void k_select(const u32* __restrict__ hist, u32* __restrict__ cutoff) {
    int lvl = threadIdx.x;
    if (lvl >= 3) return;
    const u32* h = hist + lvl * NBINS;
    u32 cum = 0, cut = 0;
    for (int b = NBINS - 1; b >= 0; --b) {
        u32 c2 = cum + h[b];
        if (c2 > CAP_) { cut = (u32)(b + 1); break; }       // can't afford this bin
        cum = c2;
        if (cum >= TOPK_) { cut = (u32)b; break; }          // superset of top-k reached
    }
    cutoff[lvl] = cut;
}

// ---------------- 3: compaction (pass 2) ----------------
__global__ void __launch_bounds__(256) k_compact(const float* __restrict__ obj,
                                                 const float* __restrict__ cls,
                                                 int n4, const u32* __restrict__ cutoffPtr,
                                                 u64* __restrict__ keys, u32* __restrict__ cnt) {
    u32 cutoff = *cutoffPtr;
    int stride = gridDim.x * blockDim.x;
    for (int i = blockIdx.x * blockDim.x + threadIdx.x; i < n4; i += stride) {
        __builtin_prefetch(reinterpret_cast<const char*>(cls) + 16ull * ((unsigned)i + (unsigned)stride), 0, 1);
        int row = i / 20;
        float os = fast_sig(obj[row]);
        float m = (os > CONF_T) ? os : 0.0f;
        float4 c = reinterpret_cast<const float4*>(cls)[i];
        float s[4];
        s[0] = m * fast_sig(c.x); s[1] = m * fast_sig(c.y);
        s[2] = m * fast_sig(c.z); s[3] = m * fast_sig(c.w);
#pragma unroll
        for (int j = 0; j < 4; ++j) {
            u32 bits = __float_as_uint(s[j]);
            if ((bits >> BIN_SHIFT) >= cutoff) {
                u32 pos = atomicAdd(cnt, 1u);
                if (pos < CAP_) {
                    u32 flat = (u32)i * 4u + (u32)j;
                    keys[pos] = ((u64)bits << 32) | (u32)(~flat);   // ties -> lowest index first
                }
            }
        }
    }
}

// ---------------- bitonic sort (descending) in LDS ----------------
template <int N, int NT>
__device__ __forceinline__ void bitonic_desc(u64* sk, int tid) {
    for (unsigned k = 2; k <= (unsigned)N; k <<= 1) {
        for (unsigned j = k >> 1; j > 0; j >>= 1) {
            for (int i = tid; i < N; i += NT) {
                int ij = i ^ (int)j;
                if (ij > i) {
                    bool up = ((i & (int)k) == 0);
                    u64 a = sk[i], b = sk[ij];
                    if ((a < b) == up) { sk[i] = b; sk[ij] = a; }
                }
            }
            __syncthreads();
        }
    }
}

// ---------------- 4: per-level sort + emit top-1000 with box decode ----------------
__global__ void __launch_bounds__(1024) k_sort_emit(const u64* __restrict__ keys,
                                                    const float* __restrict__ reg,
                                                    const float* __restrict__ anchors,
                                                    int M, float stride, int base,
                                                    float* __restrict__ candScore,
                                                    float* __restrict__ candLabel,
                                                    float* __restrict__ candBox) {
    __shared__ u64 sk[CAP_];
    int tid = threadIdx.x;
#ifdef HAVE_ASYNC_LDS
    // stage the 64KB key block into LDS via CDNA5 async DMA (ASYNCcnt-tracked)
    for (int i = tid; i < CAP_; i += 1024)
        __builtin_amdgcn_global_load_async_to_lds_b64(GV2(keys + i), LV2(&sk[i]), 0, 0);
    __builtin_amdgcn_s_wait_asynccnt(0);
#else
    for (int i = tid; i < CAP_; i += 1024) sk[i] = keys[i];
#endif
    __syncthreads();
    bitonic_desc<CAP_, 1024>(sk, tid);
    if (tid < TOPK_) {
        u64 key = sk[tid];
        u32 sb  = (u32)(key >> 32);
        u32 idx = ~(u32)key;
        float sc = __uint_as_float(sb);
        float b0 = 0.f, b1 = 0.f, b2 = 0.f, b3 = 0.f, lab = 0.f;
        if (sc > 0.0f && idx < (u32)M * (u32)NCLS_) {
            int a = (int)(idx / NCLS_);
            int c = (int)(idx % NCLS_);
            lab = (float)c;
            float4 rg = reinterpret_cast<const float4*>(reg)[a];
            float4 an = reinterpret_cast<const float4*>(anchors)[a];
            float cx = (ref_sig(rg.x) * 2.0f - 0.5f + an.x) * stride;
            float cy = (ref_sig(rg.y) * 2.0f - 0.5f + an.y) * stride;
            float w  = expf(rg.z) * an.z;
            float h  = expf(rg.w) * an.w;
            b0 = cx - 0.5f * w; b1 = cy - 0.5f * h;
            b2 = cx + 0.5f * w; b3 = cy + 0.5f * h;
        } else {
            sc = 0.0f;
        }
        int g = base + tid;
        candScore[g] = sc; candLabel[g] = lab;
        candBox[g * 4 + 0] = b0; candBox[g * 4 + 1] = b1;
        candBox[g * 4 + 2] = b2; candBox[g * 4 + 3] = b3;
    }
}

// ---------------- 5: global stable sort of 3000 + emit sorted outputs ----------------
__global__ void __launch_bounds__(1024) k_sort_global(const float* __restrict__ candScore,
                                                      const float* __restrict__ candLabel,
                                                      const float* __restrict__ candBox,
                                                      float* __restrict__ dout,
                                                      float* __restrict__ obBox,
                                                      float* __restrict__ obArea) {
    __shared__ u64 sk[4096];
    int tid = threadIdx.x;
    for (int i = tid; i < 4096; i += 1024) {
        u64 key = 0ull;
        if (i < NCAND) {
            u32 bits = __float_as_uint(candScore[i]);
            key = ((u64)bits << 32) | (u32)(~(u32)i);       // stable: lower position first
        }
        sk[i] = key;
    }
    __syncthreads();
    bitonic_desc<4096, 1024>(sk, tid);
    for (int r = tid; r < NCAND; r += 1024) {
        u64 key = sk[r];
        u32 g = ~(u32)key;                                  // < 3000 for all real keys
        float sc  = __uint_as_float((u32)(key >> 32));
        float lab = candLabel[g];
        float b0 = candBox[g * 4 + 0], b1 = candBox[g * 4 + 1];
        float b2 = candBox[g * 4 + 2], b3 = candBox[g * 4 + 3];
        dout[r * 4 + 0] = b0; dout[r * 4 + 1] = b1;
        dout[r * 4 + 2] = b2; dout[r * 4 + 3] = b3;
        dout[12000 + r] = sc;
        dout[15000 + r] = lab;
        float off = lab * 1.0e5f;                           // class-aware NMS offset (like ref)
        float o0 = b0 + off, o1 = b1 + off, o2 = b2 + off, o3 = b3 + off;
        obBox[r * 4 + 0] = o0; obBox[r * 4 + 1] = o1;
        obBox[r * 4 + 2] = o2; obBox[r * 4 + 3] = o3;
        float w = fmaxf(o2 - o0, 0.0f), h = fmaxf(o3 - o1, 0.0f);
        obArea[r] = w * h;                                  // area of OFFSET boxes, like ref
    }
}

// ---------------- 6: pairwise IoU suppression bitmask (WMMA area-sum) ----------------
// One wave per 16-row stripe. V_WMMA_F32_16X16X4_F32 computes the 16x16 tile of
// (area_i + area_j) via A=[area_i,1,0,0] rows x B=[1,area_j,0,0] cols.
// iou > 0.5  <=>  2*inter > (asum - inter + 1e-9)   (denominator strictly positive)
__global__ void __launch_bounds__(32) k_mask(const float* __restrict__ obBox,
                                             const float* __restrict__ obArea,
                                             u32* __restrict__ mask,
                                             u32* __restrict__ rowAny) {
    __shared__ u32   lmask[16 * NWORDS];
    __shared__ float rbox[16][4];
    __shared__ float rarea[16];
    const int l  = threadIdx.x;
    const int r0 = blockIdx.x * 16;
    for (int w = l; w < 16 * NWORDS; w += 32) lmask[w] = 0u;
    if (l < 16) {
        int rr = imin(r0 + l, NCAND - 1);
        float4 rb = reinterpret_cast<const float4*>(obBox)[rr];
        rbox[l][0] = rb.x; rbox[l][1] = rb.y; rbox[l][2] = rb.z; rbox[l][3] = rb.w;
        rarea[l] = obArea[rr];
    }
    __syncthreads();

    const int hi = (l >= 16) ? 8 : 0;
    // this lane's 8 fixed rows, hoisted to registers for all 188 column tiles
    float rx0[8], ry0[8], rx1[8], ry1[8], rar[8];
#pragma unroll
    for (int v = 0; v < 8; ++v) {
        int m = v + hi;
        rx0[v] = rbox[m][0]; ry0[v] = rbox[m][1];
        rx1[v] = rbox[m][2]; ry1[v] = rbox[m][3];
        rar[v] = rarea[m];
    }

#ifdef HAVE_WMMA_F32X4
    v2f A;                                   // A 16x4: lanes0-15 K0/K1, lanes16-31 K2/K3
    A.x = (l < 16) ? rarea[l] : 0.0f;        // K0 = area_row, K2 = 0
    A.y = (l < 16) ? 1.0f     : 0.0f;        // K1 = 1,        K3 = 0
#endif
    for (int j0 = 0; j0 < NCAND; j0 += 16) {
        int nloc = l & 15;
        int cIdx = j0 + nloc;
        int cc   = imin(cIdx, NCAND - 1);
        float4 cb = reinterpret_cast<const float4*>(obBox)[cc];
        float carea = obArea[cc];
#ifdef HAVE_WMMA_F32X4
        v2f B;                               // B 4x16: lanes0-15 K0/K1, lanes16-31 K2/K3
        B.x = (l < 16) ? 1.0f  : 0.0f;       // row K0 = ones
        B.y = (l < 16) ? carea : 0.0f;       // row K1 = area_col
        v8f C = {};
        v8f D = __builtin_amdgcn_wmma_f32_16x16x4_f32(false, A, false, B,
                                                      (short)0, C, false, false);
#endif
#pragma unroll
        for (int v = 0; v < 8; ++v) {
            float w = fmaxf(fminf(rx1[v], cb.z) - fmaxf(rx0[v], cb.x), 0.0f);
            float h = fmaxf(fminf(ry1[v], cb.w) - fmaxf(ry0[v], cb.y), 0.0f);
            float inter = w * h;
#ifdef HAVE_WMMA_F32X4
            float asum = D[v];
#else
            float asum = rar[v] + carea;
#endif
            float den = asum - inter + 1e-9f;                 // > 0 always
            bool  hit = (inter + inter) > den;                // iou > 0.5
            int rg = r0 + v + hi;
            if (hit && cIdx > rg && cIdx < NCAND && rg < NCAND)
                atomicOr(&lmask[(v + hi) * NWORDS + (cIdx >> 5)], 1u << (cIdx & 31));
        }
    }
    __syncthreads();
    for (int w = l; w < 16 * NWORDS; w += 32)
        mask[(size_t)(r0 + w / NWORDS) * NWORDS + (w % NWORDS)] = lmask[w];
    if (l < 16) {
        u32 o = 0;
        for (int w = 0; w < NWORDS; ++w) o |= lmask[l * NWORDS + w];
        rowAny[r0 + l] = (o != 0u) ? 1u : 0u;
    }
}

// ---------------- 7: greedy sequential NMS scan (single wave32) ----------------
// Active-row mask rows are staged through a 4-deep LDS ring with CDNA5 async
// global->LDS DMA, hiding L2 latency behind s_wait_asynccnt pipelining.
__global__ void __launch_bounds__(32) k_nms(const u32* __restrict__ mask,
                                            const u32* __restrict__ rowAny,
                                            const float* __restrict__ scores,
                                            float* __restrict__ keepOut) {
    __shared__ u32 removed[96];
    __shared__ unsigned short list[NCAND];
    __shared__ u32 acount;
#ifdef HAVE_ASYNC_LDS
    __shared__ u32 slots[4][96];             // 376B rows, b128-loaded as 384B (pad rows exist)
#endif
    const int l = threadIdx.x;
    // init removed: invalid (score<=0) or out-of-range
    for (int w = l; w < 96; w += 32) {
        u32 word = 0u;
        for (int b = 0; b < 32; ++b) {
            int j = w * 32 + b;
            bool bad = (j >= NCAND) || !(scores[j] > 0.0f);
            if (bad) word |= 1u << b;
        }
        removed[w] = word;
    }
    if (l == 0) acount = 0u;
    __syncthreads();
    // ordered compaction of rows that can suppress anything
    for (int base = 0; base < NCAND; base += 32) {
        int i = base + l;
        bool act = (i < NCAND) && (rowAny[i] != 0u);
        unsigned long long bal = __ballot(act);
        u32 m32 = (u32)bal;                                  // wave32: low 32 bits
        u32 below = (l == 0) ? 0u : (m32 & ((1u << l) - 1u));
        u32 cbase = acount;
        if (act) list[cbase + __popc(below)] = (unsigned short)i;
        __syncthreads();
        if (l == 0) acount += __popc(m32);
        __syncthreads();
    }
    u32 n = acount;
    __syncthreads();
    volatile u32* rm = removed;
#ifdef HAVE_ASYNC_LDS
    for (u32 p = 0; p < 3u && p < n; ++p) {                  // prologue: rows 0..2
        int ip = (int)list[p];
        if (l < 24)
            __builtin_amdgcn_global_load_async_to_lds_b128(
                GV4(mask + (size_t)ip * NWORDS + l * 4),
                LV4(&slots[p][l * 4]), 0, 0);
    }
    for (u32 k = 0; k < n; ++k) {
        if (k + 3u < n) {
            int ip = (int)list[k + 3u];
            if (l < 24)
                __builtin_amdgcn_global_load_async_to_lds_b128(
                    GV4(mask + (size_t)ip * NWORDS + l * 4),
                    LV4(&slots[(k + 3u) & 3u][l * 4]), 0, 0);
        }
        __builtin_amdgcn_s_wait_asynccnt(3);                 // row k (in-order) is resident
        int i = (int)list[k];
        const u32* slot = slots[k & 3u];
        u32 w0 = slot[l];
        u32 w1 = slot[32 + l];
        u32 w2 = (l < NWORDS - 64) ? slot[64 + l] : 0u;
        __syncthreads();
        bool alive = ((rm[i >> 5] >> (i & 31)) & 1u) == 0u;  // keep[i] at time i
        if (alive) {
            atomicOr(&removed[l], w0);
            atomicOr(&removed[32 + l], w1);
            if (l < NWORDS - 64) atomicOr(&removed[64 + l], w2);
        }
        __syncthreads();
    }
#else
    for (u32 k = 0; k < n; ++k) {
        int i = (int)list[k];
        if (k + 4u < n)
            __builtin_prefetch(&mask[(size_t)list[k + 4] * NWORDS + l], 0, 1);
        u32 w0 = mask[(size_t)i * NWORDS + l];
        u32 w1 = mask[(size_t)i * NWORDS + 32 + l];
        u32 w2 = (l < NWORDS - 64) ? mask[(size_t)i * NWORDS + 64 + l] : 0u;
        __syncthreads();
        bool alive = ((rm[i >> 5] >> (i & 31)) & 1u) == 0u;
        if (alive) {
            atomicOr(&removed[l], w0);
            atomicOr(&removed[32 + l], w1);
            if (l < NWORDS - 64) atomicOr(&removed[64 + l], w2);
        }
        __syncthreads();
    }
#endif
    __syncthreads();
    for (int j = l; j < NCAND; j += 32)
        keepOut[j] = ((removed[j >> 5] >> (j & 31)) & 1u) ? 0.0f : 1.0f;
}

// ---------------- launcher ----------------
extern "C" void kernel_launch(void* const* d_in, const int* in_sizes, int n_in,
                              void* d_out, int out_size, void* d_ws, size_t ws_size,
                              hipStream_t stream) {
    (void)n_in; (void)out_size; (void)ws_size;
    uint8_t* ws = (uint8_t*)d_ws;
    // zeroed region: [hist | cnt | cutoff | keys], contiguous
    u32* hist   = (u32*)(ws + 0);                 // 3*8192*4   = 98304
    u32* cnt    = (u32*)(ws + 98304);             // 16 bytes
    u32* cutoff = (u32*)(ws + 98320);             // 16 bytes
    u64* keys   = (u64*)(ws + 98336);             // 3*8192*8   = 196608 -> end 294944
    const int zeroWords = (98304 + 16 + 16 + 196608) / 4;
    size_t o = 295168;                            // 256-aligned cursor
    float* candScore = (float*)(ws + o); o += 12032;
    float* candLabel = (float*)(ws + o); o += 12032;
    float* candBox   = (float*)(ws + o); o += 48128;
    float* obBox     = (float*)(ws + o); o += 48128;
    float* obArea    = (float*)(ws + o); o += 12032;
    u32*   mask      = (u32*)(ws + o);   o += (size_t)NROWPAD * NWORDS * 4; // 1,131,008
    u32*   rowAny    = (u32*)(ws + o);   o += 12032;

    float* dout = (float*)d_out;
    const float strides[3] = {8.0f, 16.0f, 32.0f};
    const int Ms[3] = {in_sizes[0], in_sizes[4], in_sizes[8]};

    k_zero<<<80, 256, 0, stream>>>((u32*)ws, zeroWords);

    for (int lvl = 0; lvl < 3; ++lvl) {
        const float* obj = (const float*)d_in[4 * lvl + 0];
        const float* cls = (const float*)d_in[4 * lvl + 1];
        int n4 = Ms[lvl] * 20;
        int blocks = (n4 + 255) / 256; if (blocks > 4096) blocks = 4096;
        k_hist<<<blocks, 256, 0, stream>>>(obj, cls, n4, hist + lvl * NBINS);
    }
    k_select<<<1, 32, 0, stream>>>(hist, cutoff);
    for (int lvl = 0; lvl < 3; ++lvl) {
        const float* obj = (const float*)d_in[4 * lvl + 0];
        const float* cls = (const float*)d_in[4 * lvl + 1];
        int n4 = Ms[lvl] * 20;
        int blocks = (n4 + 255) / 256; if (blocks > 4096) blocks = 4096;
        k_compact<<<blocks, 256, 0, stream>>>(obj, cls, n4, cutoff + lvl,
                                              keys + (size_t)lvl * CAP_, cnt + lvl);
    }
    for (int lvl = 0; lvl < 3; ++lvl) {
        const float* reg = (const float*)d_in[4 * lvl + 2];
        const float* anc = (const float*)d_in[4 * lvl + 3];
        k_sort_emit<<<1, 1024, 0, stream>>>(keys + (size_t)lvl * CAP_, reg, anc,
                                            Ms[lvl], strides[lvl], lvl * TOPK_,
                                            candScore, candLabel, candBox);
    }
    k_sort_global<<<1, 1024, 0, stream>>>(candScore, candLabel, candBox,
                                          dout, obBox, obArea);
    k_mask<<<(NCAND + 15) / 16, 32, 0, stream>>>(obBox, obArea, mask, rowAny);
    k_nms<<<1, 32, 0, stream>>>(mask, rowAny, dout + 12000, dout + 18000);
}